// ColorAwareSmoothnessLoss_13563506721096
// MI455X (gfx1250) — compile-verified
//
#include <hip/hip_runtime.h>
#include <hip/hip_bf16.h>
#include <math.h>
#include <stdint.h>

// Problem constants (from reference): B=4, C=13, N=8192, K=16
#define BB 4
#define CC 13
#define NN 8192
#define KK 16
#define CHUNK 1024            // m-points staged in LDS per block iteration
#define NCHUNK (NN / CHUNK)   // 8

typedef __attribute__((ext_vector_type(2))) float v2f;
typedef __attribute__((ext_vector_type(8))) float v8f;

// ------------------------------------------------------------------
// Kernel 1: transpose xyz/rgb to point-major, precompute |x|^2 and
// softmax(logits) into stride-16 rows (contiguous neighbor gathers).
// Also zeroes the output accumulator (thread 0).
// ------------------------------------------------------------------
__global__ void prep_kernel(const float* __restrict__ logits,
                            const float* __restrict__ xyz,
                            const float* __restrict__ rgb,
                            float* __restrict__ xx,
                            float* __restrict__ xyzT,
                            float* __restrict__ rgbT,
                            float* __restrict__ probsT,
                            float* __restrict__ out)
{
    int t = blockIdx.x * blockDim.x + threadIdx.x;
    if (t >= BB * NN) return;
    int b = t / NN, n = t % NN;

    const float* xb = xyz + (size_t)b * 3 * NN;
    float x = xb[n], y = xb[NN + n], z = xb[2 * NN + n];
    xx[t] = x * x + y * y + z * z;
    xyzT[t * 4 + 0] = x; xyzT[t * 4 + 1] = y;
    xyzT[t * 4 + 2] = z; xyzT[t * 4 + 3] = 0.f;   // K=4 pad for WMMA

    const float* rb = rgb + (size_t)b * 3 * NN;
    rgbT[t * 4 + 0] = rb[n];
    rgbT[t * 4 + 1] = rb[NN + n];
    rgbT[t * 4 + 2] = rb[2 * NN + n];
    rgbT[t * 4 + 3] = 0.f;

    const float* lb = logits + (size_t)b * CC * NN + n;
    float l[CC];
    float m = -3.4e38f;
#pragma unroll
    for (int c = 0; c < CC; ++c) { l[c] = lb[(size_t)c * NN]; m = fmaxf(m, l[c]); }
    float s = 0.f;
#pragma unroll
    for (int c = 0; c < CC; ++c) { l[c] = __expf(l[c] - m); s += l[c]; }
    float inv = 1.f / s;
#pragma unroll
    for (int c = 0; c < CC; ++c) probsT[t * 16 + c] = l[c] * inv;
    probsT[t * 16 + 13] = 0.f; probsT[t * 16 + 14] = 0.f; probsT[t * 16 + 15] = 0.f;

    if (t == 0) out[0] = 0.f;
}

// ------------------------------------------------------------------
// Kernel 2: brute-force KNN via V_WMMA_F32_16X16X4_F32 with
// double-buffered GLOBAL_LOAD_ASYNC_TO_LDS_B128 chunk staging.
//   A 16x4 f32 layout: lane<16 -> (K0,K1)=(x,y); lane>=16 -> (K2,K3)=(z,0)
//   D 16x16 f32 layout: vgpr j, lane l -> row M=j+8*(l>=16), col N=l&15
// Each lane keeps a branch-free sorted top-16 of its candidate stream;
// lanes l and l+16 (same query column) merge via shfl_xor bitonic min.
// ------------------------------------------------------------------
__global__ void __launch_bounds__(256)
knn_kernel(const float* __restrict__ xyzT,
           const float* __restrict__ xxg,
           int* __restrict__ knn)
{
    __shared__ __align__(16) float lds_xyz[2][CHUNK * 4];   // 32 KB
    __shared__ __align__(16) float lds_xx[2][CHUNK];        //  8 KB

    const int tid  = threadIdx.x;
    const int lane = tid & 31;
    const int wave = tid >> 5;
    const int b    = blockIdx.x >> 6;          // 64 blocks per batch
    const int tg   = blockIdx.x & 63;
    const int qcol = lane & 15;
    const int half = lane >> 4;
    const int nq   = (tg * 8 + wave) * 16 + qcol;  // this lane's query point
    const int k0   = half * 2;                     // K-components this lane carries

    const size_t bofs = (size_t)b * NN;

    v2f bop;                                       // B operand: 16 query points
    bop.x = xyzT[(bofs + nq) * 4 + k0];
    bop.y = xyzT[(bofs + nq) * 4 + k0 + 1];
    const float xxq = xxg[bofs + nq];

    float hd[KK]; int hi[KK];                      // sorted ascending top-K
#pragma unroll
    for (int i = 0; i < KK; ++i) { hd[i] = 3.4e38f; hi[i] = 0; }

    // Issue async DMA of one 1024-point chunk into LDS buffer `buf`.
    // Each of the 256 threads moves 16 xyzT floats (4x B128) + 4 xx floats.
    auto issue_chunk = [&](int c0, int buf) {
        const uint64_t gsrc = (uint64_t)(uintptr_t)(xyzT + (bofs + c0) * 4 + tid * 16);
        const uint32_t ldst = (uint32_t)(uintptr_t)&lds_xyz[buf][tid * 16];
#pragma unroll
        for (int q = 0; q < 4; ++q) {
            asm volatile("global_load_async_to_lds_b128 %0, %1, off"
                         :: "v"(ldst + 16u * q), "v"(gsrc + 16u * q) : "memory");
        }
        const uint64_t gxx = (uint64_t)(uintptr_t)(xxg + bofs + c0 + tid * 4);
        const uint32_t lxx = (uint32_t)(uintptr_t)&lds_xx[buf][tid * 4];
        asm volatile("global_load_async_to_lds_b128 %0, %1, off"
                     :: "v"(lxx), "v"(gxx) : "memory");
    };

    issue_chunk(0, 0);

    for (int c = 0; c < NCHUNK; ++c) {
        const int cur = c & 1;
        // Wait for this wave's outstanding DMA, then make it block-visible.
        asm volatile("s_wait_asynccnt 0x0" ::: "memory");
        __syncthreads();
        // Overlap: start DMA of the next chunk into the other buffer.
        if (c + 1 < NCHUNK) issue_chunk((c + 1) * CHUNK, cur ^ 1);

        const int c0 = c * CHUNK;
        for (int mb = 0; mb < CHUNK; mb += 16) {
            const int pa = mb + qcol;              // A operand: tile of 16 m-points
            v2f aop;
            aop.x = lds_xyz[cur][pa * 4 + k0];
            aop.y = lds_xyz[cur][pa * 4 + k0 + 1];

            v8f acc = {};
            // (neg_a, A, neg_b, B, c_mod, C, reuse_a, reuse_b)
            acc = __builtin_amdgcn_wmma_f32_16x16x4_f32(
                      false, aop, false, bop, (short)0, acc, false, false);

            const int mrow = mb + half * 8;        // D rows held by this lane
            const float4 xxa = *(const float4*)&lds_xx[cur][mrow];
            const float4 xxb = *(const float4*)&lds_xx[cur][mrow + 4];
            float xxm[8] = { xxa.x, xxa.y, xxa.z, xxa.w,
                             xxb.x, xxb.y, xxb.z, xxb.w };
#pragma unroll
            for (int j = 0; j < 8; ++j) {
                const float dist = fmaf(-2.f, acc[j], xxq + xxm[j]);
                const int   mg   = c0 + mrow + j;
                if (dist < hd[KK - 1]) {           // rare after warm-up
                    // Branch-free insert: compare/select chain, stays in VGPRs.
                    float dd = dist; int di = mg;
#pragma unroll
                    for (int i = 0; i < KK; ++i) {
                        const bool  lt = dd < hd[i];
                        const float fo = hd[i];
                        const int   io = hi[i];
                        hd[i] = lt ? dd : fo;
                        hi[i] = lt ? di : io;
                        dd = lt ? fo : dd;
                        di = lt ? io : di;
                    }
                }
            }
        }
    }

    // Merge lane l with lane l^16 (same query): min(L[k], P[15-k]) is the
    // 16-smallest of the union; loss sums over k so order is irrelevant.
    int outI[KK];
#pragma unroll
    for (int kk = 0; kk < KK; ++kk) {
        const float pd = __shfl_xor(hd[KK - 1 - kk], 16, 32);
        const int   pi = __shfl_xor(hi[KK - 1 - kk], 16, 32);
        outI[kk] = (hd[kk] <= pd) ? hi[kk] : pi;
    }
    if (half == 0) {
        int* dst = knn + (bofs + nq) * KK;
#pragma unroll
        for (int kk = 0; kk < KK; ++kk) dst[kk] = outI[kk];
    }
}

// ------------------------------------------------------------------
// Kernel 3: gather 16 neighbors per point, weight = exp(-10*|rgb diff|),
// prob_diff = L1 over 13 channels; block-reduce, atomic add of the mean.
// ------------------------------------------------------------------
__global__ void loss_kernel(const float* __restrict__ rgbT,
                            const float* __restrict__ probsT,
                            const int* __restrict__ knn,
                            float* __restrict__ out)
{
    __shared__ float red[256];
    int t = blockIdx.x * blockDim.x + threadIdx.x;
    float acc = 0.f;
    if (t < BB * NN) {
        const int b = t / NN;
        const size_t bofs = (size_t)b * NN;
        const float r0 = rgbT[t * 4 + 0], r1 = rgbT[t * 4 + 1], r2 = rgbT[t * 4 + 2];
        float p[CC];
#pragma unroll
        for (int c = 0; c < CC; ++c) p[c] = probsT[t * 16 + c];
#pragma unroll 4
        for (int k = 0; k < KK; ++k) {
            const int    j = knn[t * KK + k];
            const size_t g = bofs + (size_t)j;
            const float d0 = r0 - rgbT[g * 4 + 0];
            const float d1 = r1 - rgbT[g * 4 + 1];
            const float d2 = r2 - rgbT[g * 4 + 2];
            const float rd = sqrtf(d0 * d0 + d1 * d1 + d2 * d2);
            float pdiff = 0.f;
#pragma unroll
            for (int c = 0; c < CC; ++c) pdiff += fabsf(p[c] - probsT[g * 16 + c]);
            acc += __expf(-10.f * rd) * pdiff;
        }
    }
    red[threadIdx.x] = acc;
    __syncthreads();
    for (int s = 128; s > 0; s >>= 1) {
        if (threadIdx.x < s) red[threadIdx.x] += red[threadIdx.x + s];
        __syncthreads();
    }
    if (threadIdx.x == 0)
        atomicAdd(out, red[0] * (1.f / (float)(BB * NN * KK)));
}

// ------------------------------------------------------------------
extern "C" void kernel_launch(void* const* d_in, const int* in_sizes, int n_in,
                              void* d_out, int out_size, void* d_ws, size_t ws_size,
                              hipStream_t stream) {
    const float* logits = (const float*)d_in[0];
    const float* xyz    = (const float*)d_in[1];
    const float* rgb    = (const float*)d_in[2];
    float* out = (float*)d_out;

    const int BN = BB * NN;                // 32768
    float* ws     = (float*)d_ws;
    float* xx     = ws;                    // BN floats
    float* xyzT   = ws + BN;               // BN*4 floats (x,y,z,0)
    float* rgbT   = ws + (size_t)BN * 5;   // BN*4 floats
    float* probsT = ws + (size_t)BN * 9;   // BN*16 floats (13 used)
    int*   knn    = (int*)(ws + (size_t)BN * 25);  // BN*K ints
    (void)in_sizes; (void)n_in; (void)out_size; (void)ws_size;

    prep_kernel<<<(BN + 255) / 256, 256, 0, stream>>>(
        logits, xyz, rgb, xx, xyzT, rgbT, probsT, out);

    knn_kernel<<<BB * 64, 256, 0, stream>>>(xyzT, xx, knn);

    loss_kernel<<<(BN + 255) / 256, 256, 0, stream>>>(rgbT, probsT, knn, out);
}